// MultiFusion_64304250355745
// MI455X (gfx1250) — compile-verified
//
#include <hip/hip_runtime.h>
#include <stdint.h>

// Problem shape (fixed by the reference): B=32, N=2048, D=64, C=8.
#define BSZ   32
#define NN    2048
#define DD    64
#define CC    8
#define ROWK  512      // D*C floats per (b,n) row
#define ROWPAD 544     // 512 dwords + 1 pad dword per 16 dwords (TDM pad_enable)
#define ROWS_PER_BLOCK 8
#define THREADS 256

typedef unsigned int uint32;
typedef uint32 v4u __attribute__((ext_vector_type(4)));
typedef int    v4i __attribute__((ext_vector_type(4)));
typedef int    v8i __attribute__((ext_vector_type(8)));

__global__ __launch_bounds__(THREADS)
void multifusion_kernel(const float* __restrict__ Xs,
                        const float* __restrict__ W,
                        float* __restrict__ out) {
    __shared__ float lds[ROWS_PER_BLOCK * ROWPAD];

    const int tid  = threadIdx.x;
    const int wave = tid >> 5;
    const int lane = tid & 31;
    const long long blockRow0 = (long long)blockIdx.x * ROWS_PER_BLOCK;

    // ---------------- Stage the 8-row tile (16 KB) into LDS ----------------
#if __has_builtin(__builtin_amdgcn_tensor_load_to_lds)
    if (tid == 0) {
        // Tensor DMA descriptor (D#), 2D: tile = 8 rows x 512 f32, with LDS
        // padding of 1 dword every 16 dwords (pad_interval=3 -> 16dw, amount=0 -> 1dw)
        unsigned long long gaddr =
            (unsigned long long)(const void*)Xs +
            (unsigned long long)blockRow0 * (ROWK * 4ull);
        unsigned int ldsaddr = (unsigned int)(uintptr_t)(void*)lds;

        v4u g0;
        g0.x = 1u;                                            // count=1 (valid), user mode
        g0.y = ldsaddr;                                       // lds_addr (bytes)
        g0.z = (uint32)(gaddr & 0xFFFFFFFFull);               // global_addr[31:0]
        g0.w = (uint32)((gaddr >> 32) & 0x01FFFFFFull)        // global_addr[56:32]
             | 0x80000000u;                                   // type=2 ("image")

        v8i g1;
        g1[0] = (int)((2u << 16)      // data_size = 4B
                    | (1u << 20)      // pad_enable
                    | (3u << 22));    // pad_interval: 16 dwords (pad_amount=0 -> 1 dword)
        g1[1] = (int)((uint32)ROWK << 16);          // tensor_dim0[15:0] = 512
        g1[2] = (int)((uint32)ROWS_PER_BLOCK << 16);// dim0 hi=0 | tensor_dim1 lo16 = 8
        g1[3] = (int)((uint32)ROWK << 16);          // dim1 hi16=0 | tile_dim0 = 512
        g1[4] = ROWS_PER_BLOCK;                     // tile_dim1 = 8, tile_dim2 = 0
        g1[5] = ROWK;                               // tensor_dim0_stride lo32 = 512
        g1[6] = 0;
        g1[7] = 0;

        v4i gz = {0, 0, 0, 0};
#if defined(__clang_major__) && (__clang_major__ >= 23)
        v8i gz8 = {0, 0, 0, 0, 0, 0, 0, 0};
        __builtin_amdgcn_tensor_load_to_lds(g0, g1, gz, gz, gz8, 0);
#else
        __builtin_amdgcn_tensor_load_to_lds(g0, g1, gz, gz, 0);
#endif
        __builtin_amdgcn_s_wait_tensorcnt(0);
    }
    // Make sure the compiler treats LDS as written by the TDM op.
    __asm__ volatile("" : : "r"(&lds[0]) : "memory");
#else
    // Fallback: cooperative copy replicating the padded LDS layout.
    for (int idx = tid; idx < ROWS_PER_BLOCK * ROWK; idx += THREADS) {
        int r = idx >> 9, k = idx & (ROWK - 1);
        lds[r * ROWPAD + k + (k >> 4)] = Xs[(blockRow0 + r) * ROWK + k];
    }
#endif
    __syncthreads();

    // ---------------- One wave per (b,n) row ----------------
    const long long gRow = blockRow0 + wave;
    const float* ldsrow = &lds[wave * ROWPAD];

    // Pass 1: logits. Channel-per-lane layout: flat k = lane + 32*i
    //   -> c = lane%8 (fixed), d = lane/8 + 4*i.  Padded dword = base2 + 34*i.
    const int base2 = lane + (lane >> 4);
    float p = 0.0f;
#pragma unroll
    for (int i = 0; i < 16; ++i) {
        float x = ldsrow[base2 + 34 * i];
        float w = W[lane + 32 * i];           // same (d*8+c) flattening; L2/WGP$-resident
        p = fmaf(x, w, p);
    }
    // Reduce over d: lanes sharing lane%8 (xor 8, 16).
    p += __shfl_xor(p, 8, 32);
    p += __shfl_xor(p, 16, 32);
    // Softmax over the 8 channels (xor 1,2,4 spans each group of 8 lanes).
    float m = p;
    m = fmaxf(m, __shfl_xor(m, 1, 32));
    m = fmaxf(m, __shfl_xor(m, 2, 32));
    m = fmaxf(m, __shfl_xor(m, 4, 32));
    float e = __expf(p - m);
    float s = e;
    s += __shfl_xor(s, 1, 32);
    s += __shfl_xor(s, 2, 32);
    s += __shfl_xor(s, 4, 32);
    const float gate = e / s;                 // lane L holds gate[L%8]

    // Broadcast all 8 gates to every lane (lane c holds gate[c]).
    float g[8];
#pragma unroll
    for (int c = 0; c < 8; ++c) g[c] = __shfl(gate, c, 32);

    // Pass 2: combine. Row-major per-lane layout: k = 16*lane + j
    //   -> padded dword = 17*lane + j (conflict-free: gcd(17,64)=1).
    //   Lane covers d = 2*lane and 2*lane+1, all 8 channels each.
    const float* xa = ldsrow + 17 * lane;
    float o0 = 0.0f, o1 = 0.0f;
#pragma unroll
    for (int c = 0; c < 8; ++c) {
        o0 = fmaf(xa[c],     g[c], o0);
        o1 = fmaf(xa[8 + c], g[c], o1);
    }
    float2* orow = (float2*)(out + gRow * DD);
    orow[lane] = make_float2(o0, o1);         // wave writes one coalesced 256B row
}

extern "C" void kernel_launch(void* const* d_in, const int* in_sizes, int n_in,
                              void* d_out, int out_size, void* d_ws, size_t ws_size,
                              hipStream_t stream) {
    const float* Xs = (const float*)d_in[0];  // [B,N,D,C] f32
    const float* W  = (const float*)d_in[1];  // [D,C]     f32
    float* out = (float*)d_out;               // [B,N,D]   f32

    const int rows = BSZ * NN;                              // 65536
    dim3 grid(rows / ROWS_PER_BLOCK);                       // 8192 blocks
    multifusion_kernel<<<grid, THREADS, 0, stream>>>(Xs, W, out);
}